// GraphEditModel_39900246180429
// MI455X (gfx1250) — compile-verified
//
#include <hip/hip_runtime.h>
#include <hip/hip_bf16.h>

typedef __attribute__((ext_vector_type(2))) float v2f;
typedef __attribute__((ext_vector_type(4))) float v4f;
typedef __attribute__((ext_vector_type(8))) float v8f;

#define NN 100000
#define DD 256
#define EE 262144
#define CC 5
#define LDA_PAD 260   // 260 % 64 == 4 -> v2f A-loads cover all 64 LDS banks

// One 16x16 fp32 output tile, K=256, A strip in LDS (row stride LDA), B slab in
// LDS as [k][n] row-major 16 wide. Emits 64x v_wmma_f32_16x16x4_f32.
template <int LDA>
__device__ __forceinline__ v8f wmma_k256(const float* __restrict__ sArow,
                                         const float* __restrict__ sB,
                                         int col, int koff) {
  v8f acc = {0.f, 0.f, 0.f, 0.f, 0.f, 0.f, 0.f, 0.f};
  const float* ap = sArow + col * LDA + koff;   // A: row = lane&15, K pair at koff
  const float* bp = sB + koff * 16 + col;       // B: col = lane&15, K pair at koff
#pragma unroll 8
  for (int k = 0; k < 256; k += 4) {
    v2f a = *(const v2f*)(ap + k);
    v2f b;
    b.x = bp[k * 16];        // B[k+koff  ][col]
    b.y = bp[k * 16 + 16];   // B[k+koff+1][col]
    acc = __builtin_amdgcn_wmma_f32_16x16x4_f32(
        /*neg_a=*/false, a, /*neg_b=*/false, b,
        /*c_mod=*/(short)0, acc, /*reuse_a=*/false, /*reuse_b=*/false);
  }
  return acc;
}

// Register-prefetch helpers for the 256x16 weight slab (4096 floats, 128 thr).
__device__ __forceinline__ void slab_fetch(const float* __restrict__ W, int nt,
                                           int tid, float* __restrict__ wreg) {
#pragma unroll
  for (int j = 0; j < 32; ++j) {
    const int i = tid + j * 128;                 // (i&15) const, (i>>4) = tid>>4 + 8j
    wreg[j] = W[(i >> 4) * DD + nt * 16 + (i & 15)];
  }
}
__device__ __forceinline__ void slab_commit(float* __restrict__ sB, int tid,
                                            const float* __restrict__ wreg) {
#pragma unroll
  for (int j = 0; j < 32; ++j) sB[tid + j * 128] = wreg[j];
}

// ---- Ws = W_agg[0:256,:] + W_agg[256:512,:],  b2 = 2*b_agg ----
__global__ void prep_kernel(const float* __restrict__ Wagg,
                            const float* __restrict__ bagg,
                            float* __restrict__ Ws, float* __restrict__ b2) {
  int i = blockIdx.x * blockDim.x + threadIdx.x;
  if (i < DD * DD) Ws[i] = Wagg[i] + Wagg[i + DD * DD];
  if (i < DD) b2[i] = 2.0f * bagg[i];
}

// ---- Edge path: 64 edges / block (4 waves x 16 rows) ----
__global__ __launch_bounds__(128) void edge_kernel(
    const float* __restrict__ feat, const int* __restrict__ pair,
    const float* __restrict__ Ws, const float* __restrict__ b2,
    const float* __restrict__ We1, const float* __restrict__ be1,
    const float* __restrict__ We2, const float* __restrict__ be2,
    float* __restrict__ outE) {
  __shared__ __align__(16) float sA[64 * LDA_PAD];  // s = feat[src]+feat[dst]
  __shared__ __align__(16) float sE[64 * LDA_PAD];  // ed_feat
  __shared__ __align__(16) float sB[256 * 16];      // weight column slab

  const int tid  = threadIdx.x;
  const int lane = tid & 31;
  const int wave = tid >> 5;
  const int col  = lane & 15;
  const int koff = (lane >> 4) * 2;
  const int rbase = (lane >> 4) * 8;
  const long long e0 = (long long)blockIdx.x * 64;

  float wreg[32];
  slab_fetch(Ws, 0, tid, wreg);  // prefetch first slab while gathering

  // Phase 1: gather + add. 2 threads per edge row, float4 traffic.
  {
    const int lr = tid >> 1, half = tid & 1;
    const long long e = e0 + lr;
    const int si = pair[2 * e];
    const int di = pair[2 * e + 1];
    const float* ps = feat + (size_t)si * DD + half * 128;
    const float* pd = feat + (size_t)di * DD + half * 128;
    float* dp = sA + lr * LDA_PAD + half * 128;
#pragma unroll
    for (int q = 0; q < 32; ++q) {
      v4f a = *(const v4f*)(ps + 4 * q);
      v4f b = *(const v4f*)(pd + 4 * q);
      *(v4f*)(dp + 4 * q) = a + b;
    }
  }

  // Phase 2: ed = s @ Ws + 2*b_agg (weight slab shared by 4 waves; slab for
  // tile nt+1 is fetched to registers while tile nt is consumed by WMMAs).
#pragma unroll 1
  for (int nt = 0; nt < 16; ++nt) {
    __syncthreads();                       // previous slab fully consumed
    slab_commit(sB, tid, wreg);
    __syncthreads();                       // slab ready
    if (nt < 15) slab_fetch(Ws, nt + 1, tid, wreg);   // overlaps WMMA phase
    v8f acc = wmma_k256<LDA_PAD>(sA + wave * 16 * LDA_PAD, sB, col, koff);
    const float bb = b2[nt * 16 + col];
    float* ep = sE + (wave * 16 + rbase) * LDA_PAD + nt * 16 + col;
#pragma unroll
    for (int r = 0; r < 8; ++r) ep[r * LDA_PAD] = acc[r] + bb;
  }

  // Phase 3: h = relu(ed @ We1 + be1); fused pred = h @ We2 in registers.
  float acc2[8][CC];
#pragma unroll
  for (int r = 0; r < 8; ++r)
#pragma unroll
    for (int c = 0; c < CC; ++c) acc2[r][c] = 0.f;

  slab_fetch(We1, 0, tid, wreg);
#pragma unroll 1
  for (int nt = 0; nt < 16; ++nt) {
    __syncthreads();
    slab_commit(sB, tid, wreg);
    __syncthreads();
    if (nt < 15) slab_fetch(We1, nt + 1, tid, wreg);
    v8f h = wmma_k256<LDA_PAD>(sE + wave * 16 * LDA_PAD, sB, col, koff);
    const float bb = be1[nt * 16 + col];
    float w2[CC];
#pragma unroll
    for (int c = 0; c < CC; ++c) w2[c] = We2[(nt * 16 + col) * CC + c];
#pragma unroll
    for (int r = 0; r < 8; ++r) {
      float hv = fmaxf(h[r] + bb, 0.0f);
#pragma unroll
      for (int c = 0; c < CC; ++c) acc2[r][c] = fmaf(hv, w2[c], acc2[r][c]);
    }
  }

  // Butterfly K-reduction over the 16 lanes sharing a row group.
#pragma unroll
  for (int r = 0; r < 8; ++r)
#pragma unroll
    for (int c = 0; c < CC; ++c) {
      float v = acc2[r][c];
      v += __shfl_xor(v, 8, 16);
      v += __shfl_xor(v, 4, 16);
      v += __shfl_xor(v, 2, 16);
      v += __shfl_xor(v, 1, 16);
      acc2[r][c] = v;
    }
#pragma unroll
  for (int c = 0; c < CC; ++c)
    if (col == c) {
#pragma unroll
      for (int r = 0; r < 8; ++r) {
        long long e = e0 + wave * 16 + rbase + r;
        outE[e * CC + c] = acc2[r][c] + be2[c];
      }
    }
}

// ---- Node path: relu(nf@Wn1+bn1)@Wn2+bn2, 64 nodes / block ----
__global__ __launch_bounds__(128) void node_kernel(
    const float* __restrict__ feat,
    const float* __restrict__ Wn1, const float* __restrict__ bn1,
    const float* __restrict__ Wn2, const float* __restrict__ bn2,
    float* __restrict__ outN) {
  __shared__ __align__(16) float sA[64 * LDA_PAD];
  __shared__ __align__(16) float sB[256 * 16];

  const int tid  = threadIdx.x;
  const int lane = tid & 31;
  const int wave = tid >> 5;
  const int col  = lane & 15;
  const int koff = (lane >> 4) * 2;
  const int rbase = (lane >> 4) * 8;
  const long long n0 = (long long)blockIdx.x * 64;

  float wreg[32];
  slab_fetch(Wn1, 0, tid, wreg);

  {
    const int lr = tid >> 1, half = tid & 1;
    const long long n = n0 + lr;
    float* dp = sA + lr * LDA_PAD + half * 128;
    if (n < NN) {
      const float* ps = feat + (size_t)n * DD + half * 128;
#pragma unroll
      for (int q = 0; q < 32; ++q) *(v4f*)(dp + 4 * q) = *(const v4f*)(ps + 4 * q);
    } else {
      v4f z = {0.f, 0.f, 0.f, 0.f};
#pragma unroll
      for (int q = 0; q < 32; ++q) *(v4f*)(dp + 4 * q) = z;
    }
  }

  float acc2[8][2];
#pragma unroll
  for (int r = 0; r < 8; ++r) { acc2[r][0] = 0.f; acc2[r][1] = 0.f; }

#pragma unroll 1
  for (int nt = 0; nt < 16; ++nt) {
    __syncthreads();
    slab_commit(sB, tid, wreg);
    __syncthreads();
    if (nt < 15) slab_fetch(Wn1, nt + 1, tid, wreg);
    v8f h = wmma_k256<LDA_PAD>(sA + wave * 16 * LDA_PAD, sB, col, koff);
    const float bb = bn1[nt * 16 + col];
    float w0 = Wn2[(nt * 16 + col) * 2 + 0];
    float w1 = Wn2[(nt * 16 + col) * 2 + 1];
#pragma unroll
    for (int r = 0; r < 8; ++r) {
      float hv = fmaxf(h[r] + bb, 0.0f);
      acc2[r][0] = fmaf(hv, w0, acc2[r][0]);
      acc2[r][1] = fmaf(hv, w1, acc2[r][1]);
    }
  }

#pragma unroll
  for (int r = 0; r < 8; ++r)
#pragma unroll
    for (int c = 0; c < 2; ++c) {
      float v = acc2[r][c];
      v += __shfl_xor(v, 8, 16);
      v += __shfl_xor(v, 4, 16);
      v += __shfl_xor(v, 2, 16);
      v += __shfl_xor(v, 1, 16);
      acc2[r][c] = v;
    }
#pragma unroll
  for (int c = 0; c < 2; ++c)
    if (col == c) {
#pragma unroll
      for (int r = 0; r < 8; ++r) {
        long long n = n0 + wave * 16 + rbase + r;
        if (n < NN) outN[n * 2 + c] = acc2[r][c] + bn2[c];
      }
    }
}

extern "C" void kernel_launch(void* const* d_in, const int* in_sizes, int n_in,
                              void* d_out, int out_size, void* d_ws, size_t ws_size,
                              hipStream_t stream) {
  const float* node_feat = (const float*)d_in[0];
  const int*   pair      = (const int*)d_in[1];
  const float* W_agg     = (const float*)d_in[2];
  const float* b_agg     = (const float*)d_in[3];
  const float* We1       = (const float*)d_in[4];
  const float* be1       = (const float*)d_in[5];
  const float* We2       = (const float*)d_in[6];
  const float* be2       = (const float*)d_in[7];
  const float* Wn1       = (const float*)d_in[8];
  const float* bn1       = (const float*)d_in[9];
  const float* Wn2       = (const float*)d_in[10];
  const float* bn2       = (const float*)d_in[11];

  float* outN = (float*)d_out;                 // (N,2)
  float* outE = outN + (size_t)NN * 2;         // (E,5)
  float* Ws = (float*)d_ws;                    // 256*256
  float* b2 = Ws + DD * DD;                    // 256

  prep_kernel<<<(DD * DD + 255) / 256, 256, 0, stream>>>(W_agg, b_agg, Ws, b2);
  node_kernel<<<(NN + 63) / 64, 128, 0, stream>>>(node_feat, Wn1, bn1, Wn2, bn2, outN);
  edge_kernel<<<EE / 64, 128, 0, stream>>>(node_feat, pair, Ws, b2,
                                           We1, be1, We2, be2, outE);
}